// EmbedMatcher_62912680952547
// MI455X (gfx1250) — compile-verified
//
#include <hip/hip_runtime.h>
#include <hip/hip_bf16.h>
#include <math.h>

// ---------------- problem constants (match reference) ----------------
#define Bq    4096
#define FEW   5
#define NB    64
#define Dd    128          // D
#define DM    256          // 2*D
#define DI    512          // 2*DM
#define Hh    512          // H
#define NG    2048         // 4*H
#define NSTEPS 4
#define LNEPS 1e-6f

typedef __attribute__((ext_vector_type(2))) float v2f;
typedef __attribute__((ext_vector_type(8))) float v8f;

// ---------------------------------------------------------------------
// gfx1250 async global->LDS staging (ASYNCcnt-tracked) with safe fallback.
// Builtin signature (from clang diagnostic): param0 is a NON-const pointer
// to 'int __attribute__((vector_size(8)))' in the global (__device__/AS1)
// address space; we mirror that for the LDS (AS3) destination.
// ---------------------------------------------------------------------
#if __has_builtin(__builtin_amdgcn_global_load_async_to_lds_b64)
#define ASYNC_LDS_COPY 1
#endif

typedef int b64vec __attribute__((vector_size(8)));
typedef __attribute__((address_space(1))) b64vec gb64vec;
typedef __attribute__((address_space(3))) b64vec lb64vec;

__device__ __forceinline__ void copy_f2_to_lds(const float* g, float* l) {
#ifdef ASYNC_LDS_COPY
    __builtin_amdgcn_global_load_async_to_lds_b64(
        (gb64vec*)g, (lb64vec*)l, 0, 0);
#else
    *(float2*)l = *(const float2*)g;
#endif
}

__device__ __forceinline__ void async_copy_fence() {
#ifdef ASYNC_LDS_COPY
#if __has_builtin(__builtin_amdgcn_s_wait_asynccnt)
    __builtin_amdgcn_s_wait_asynccnt(0);
#else
    asm volatile("s_wait_asynccnt 0x0" ::: "memory");
#endif
#endif
}

// =====================================================================
// Kernel 1: gather-and-sum neighbor embeddings.
// One wave per (row, side); emb row = 128 f32 = 512B -> one float4 per
// lane, fully coalesced; global_prefetch_b8 for the next neighbor rows.
// =====================================================================
__global__ void gather_sum_kernel(const int* __restrict__ qlc,
                                  const int* __restrict__ qrc,
                                  const int* __restrict__ slc,
                                  const int* __restrict__ src_,
                                  const float* __restrict__ emb,
                                  float* __restrict__ Sq,
                                  float* __restrict__ Ss) {
    int wid  = blockIdx.x * (blockDim.x >> 5) + (threadIdx.x >> 5);
    int lane = threadIdx.x & 31;
    const int total = (Bq + FEW) * 2;
    if (wid >= total) return;

    int row, side;
    const int* conn;
    float* outp;
    if (wid < Bq * 2) {
        row  = wid >> 1; side = wid & 1;
        conn = (side ? qrc : qlc) + (size_t)row * NB * 2;
        outp = Sq + (size_t)row * (2 * DM) + side * DM;
    } else {
        int t = wid - Bq * 2;
        row  = t >> 1; side = t & 1;
        conn = (side ? src_ : slc) + (size_t)row * NB * 2;
        outp = Ss + (size_t)row * (2 * DM) + side * DM;
    }

    float4 ar = make_float4(0.f, 0.f, 0.f, 0.f);
    float4 ae = make_float4(0.f, 0.f, 0.f, 0.f);
    for (int k = 0; k < NB; ++k) {
        int ri = conn[2 * k];
        int ei = conn[2 * k + 1];
        if (k + 1 < NB) {
            __builtin_prefetch(emb + (size_t)conn[2 * k + 2] * Dd, 0, 1);
            __builtin_prefetch(emb + (size_t)conn[2 * k + 3] * Dd, 0, 1);
        }
        float4 r = ((const float4*)(emb + (size_t)ri * Dd))[lane];
        float4 e = ((const float4*)(emb + (size_t)ei * Dd))[lane];
        ar.x += r.x; ar.y += r.y; ar.z += r.z; ar.w += r.w;
        ae.x += e.x; ae.y += e.y; ae.z += e.z; ae.w += e.w;
    }
    ((float4*)outp)[lane]        = ar;   // cols 0..127   (rel)
    ((float4*)(outp + Dd))[lane] = ae;   // cols 128..255 (ent)
}

// =====================================================================
// Kernel 2: LDS-staged fp32 WMMA GEMM.
//   C[M,N] = act((A[M,K] @ W[N,K]^T + bias*bias_scale [+res]) [/deg])
// 256-thread (8-wave) block computes a 64x64 macro-tile.  Per 32-wide
// K-chunk, A(64x32) and W(64x32) are staged into LDS (async b64 copies
// when available), then each wave computes a 16x32 strip: one LDS A
// fragment is reused by two V_WMMA_F32_16X16X4_F32 issues per K-step.
// LDS rows padded to 34 floats -> 16 distinct banks for the strided
// b64 fragment reads.  EXEC stays all-ones (no partial tiles).
// =====================================================================
#define KC   32
#define LDP  34          // padded LDS row stride (floats)

__global__ void gemm_wmma_kernel(const float* __restrict__ A, int lda,
                                 const float* __restrict__ W, int ldw,
                                 const float* __restrict__ bias, float bias_scale,
                                 const float* __restrict__ residual,
                                 const int* __restrict__ deg,
                                 float* __restrict__ C, int ldc,
                                 int M, int N, int K, int act) {
    __shared__ float As[64 * LDP];
    __shared__ float Ws[64 * LDP];

    int nblk = N >> 6;
    int bm   = blockIdx.x / nblk;
    int bn   = blockIdx.x % nblk;
    int bm64 = bm << 6, bn64 = bn << 6;

    int lane = threadIdx.x;
    int w    = threadIdx.y;            // 0..7
    int tid  = w * 32 + lane;
    int sm   = w >> 1;                 // 16-row band 0..3
    int sn0  = (w & 1) << 5;           // 0 or 32 (two 16-col subtiles)
    int l15  = lane & 15;
    int khi2 = (lane >> 4) << 1;       // K sub-offset 0 or 2

    v8f acc0 = {0.f, 0.f, 0.f, 0.f, 0.f, 0.f, 0.f, 0.f};
    v8f acc1 = {0.f, 0.f, 0.f, 0.f, 0.f, 0.f, 0.f, 0.f};

    for (int k0 = 0; k0 < K; k0 += KC) {
        // ---- stage A and W K-chunks into LDS (1024 float2 each) ----
        for (int i = 0; i < 4; ++i) {
            int e   = i * 256 + tid;
            int row = e >> 4;
            int col = (e & 15) << 1;
            copy_f2_to_lds(A + (size_t)(bm64 + row) * lda + k0 + col,
                           As + row * LDP + col);
            copy_f2_to_lds(W + (size_t)(bn64 + row) * ldw + k0 + col,
                           Ws + row * LDP + col);
        }
        async_copy_fence();
        __syncthreads();
        // ---- 8 K-steps of 4, two WMMA per step sharing the A frag ----
        for (int kk = 0; kk < KC; kk += 4) {
            v2f a  = *(const v2f*)(As + (sm * 16 + l15) * LDP + kk + khi2);
            v2f b0 = *(const v2f*)(Ws + (sn0 + l15) * LDP + kk + khi2);
            v2f b1 = *(const v2f*)(Ws + (sn0 + 16 + l15) * LDP + kk + khi2);
            acc0 = __builtin_amdgcn_wmma_f32_16x16x4_f32(false, a, false, b0,
                                                         (short)0, acc0, false, false);
            acc1 = __builtin_amdgcn_wmma_f32_16x16x4_f32(false, a, false, b1,
                                                         (short)0, acc1, false, false);
        }
        __syncthreads();
    }

    // ---- epilogue: C layout lane = n + 16*(m>=8), vgpr i -> m = 8*(lane>>4)+i
    int mb = (lane >> 4) << 3;
    for (int sub = 0; sub < 2; ++sub) {
        v8f acc = sub ? acc1 : acc0;
        int n   = bn64 + sn0 + sub * 16 + l15;
        float bv = bias ? bias[n] * bias_scale : 0.f;
        for (int i = 0; i < 8; ++i) {
            int r   = bm64 + sm * 16 + mb + i;
            float v = acc[i] + bv;
            if (residual) v += residual[(size_t)r * ldc + n];
            if (deg) { int dv = deg[r]; if (dv < 1) dv = 1; v /= (float)dv; }
            if (act == 1)      v = fmaxf(v, 0.f);
            else if (act == 2) v = tanhf(v);
            C[(size_t)r * ldc + n] = v;
        }
    }
}

// =====================================================================
// Kernel 3: whole support path in one workgroup (FEW=5 rows, tiny).
// =====================================================================
__global__ void support_path_kernel(const float* __restrict__ Ss,
                                    const int* __restrict__ sld,
                                    const int* __restrict__ srd,
                                    const float* __restrict__ gcn_w,
                                    const float* __restrict__ gcn_b,
                                    const float* __restrict__ p1w,
                                    const float* __restrict__ p1b,
                                    const float* __restrict__ p2w,
                                    const float* __restrict__ p2b,
                                    const float* __restrict__ lng,
                                    const float* __restrict__ lnb,
                                    const float* __restrict__ w_hh,
                                    const float* __restrict__ b_hh,
                                    float* __restrict__ support_g,
                                    float* __restrict__ addvec) {
    __shared__ float sn[FEW][DM];
    __shared__ float sh[FEW][DI];
    __shared__ float so[FEW][DM];
    __shared__ float sg[DM];
    __shared__ float mu[FEW], sd[FEW];
    int tid = threadIdx.x;

    for (int idx = tid; idx < FEW * DM; idx += blockDim.x) {
        int row = idx / DM, d = idx % DM;
        int side = d >> 7, dd = d & 127;
        const float* x = Ss + (size_t)row * (2 * DM) + side * DM;
        float acc = 0.f;
        for (int k = 0; k < DM; ++k) acc += x[k] * gcn_w[(size_t)dd * DM + k];
        acc += (float)NB * gcn_b[dd];
        int dv = side ? srd[row] : sld[row];
        if (dv < 1) dv = 1;
        sn[row][d] = tanhf(acc / (float)dv);
    }
    __syncthreads();
    for (int idx = tid; idx < FEW * DI; idx += blockDim.x) {
        int row = idx / DI, j = idx % DI;
        float acc = p1b[j];
        for (int k = 0; k < DM; ++k) acc += sn[row][k] * p1w[(size_t)j * DM + k];
        sh[row][j] = fmaxf(acc, 0.f);
    }
    __syncthreads();
    for (int idx = tid; idx < FEW * DM; idx += blockDim.x) {
        int row = idx / DM, j = idx % DM;
        float acc = p2b[j] + sn[row][j];
        for (int k = 0; k < DI; ++k) acc += sh[row][k] * p2w[(size_t)j * DI + k];
        so[row][j] = acc;
    }
    __syncthreads();
    if (tid < FEW) {
        float s = 0.f;
        for (int k = 0; k < DM; ++k) s += so[tid][k];
        float m = s / (float)DM;
        float v = 0.f;
        for (int k = 0; k < DM; ++k) { float dx = so[tid][k] - m; v += dx * dx; }
        mu[tid] = m;
        sd[tid] = sqrtf(v / (float)(DM - 1));
    }
    __syncthreads();
    for (int d = tid; d < DM; d += blockDim.x) {
        float s = 0.f;
        for (int row = 0; row < FEW; ++row)
            s += lng[d] * (so[row][d] - mu[row]) / (sd[row] + LNEPS) + lnb[d];
        float g = s / (float)FEW;
        sg[d] = g;
        support_g[d] = g;
    }
    __syncthreads();
    for (int j = tid; j < NG; j += blockDim.x) {
        float acc = b_hh[j];
        const float* wr = w_hh + (size_t)j * Hh + DM;
        for (int k = 0; k < DM; ++k) acc += sg[k] * wr[k];
        addvec[j] = acc;
    }
}

// =====================================================================
// Kernel 4: row-wise LayerNorm (ddof=1), one wave per row (wave32).
// =====================================================================
__global__ void layernorm_kernel(const float* __restrict__ X,
                                 const float* __restrict__ g,
                                 const float* __restrict__ b,
                                 float* __restrict__ Y) {
    int row = blockIdx.x * blockDim.y + threadIdx.y;
    if (row >= Bq) return;
    int lane = threadIdx.x;
    const float* x = X + (size_t)row * DM;
    float s = 0.f, ss = 0.f;
    for (int j = lane; j < DM; j += 32) { float v = x[j]; s += v; ss += v * v; }
    for (int off = 16; off; off >>= 1) {
        s  += __shfl_xor(s,  off, 32);
        ss += __shfl_xor(ss, off, 32);
    }
    float m   = s / (float)DM;
    float var = (ss - (float)DM * m * m) / (float)(DM - 1);
    float sdv = sqrtf(fmaxf(var, 0.f));
    float* y = Y + (size_t)row * DM;
    for (int j = lane; j < DM; j += 32)
        y[j] = g[j] * (x[j] - m) / (sdv + LNEPS) + b[j];
}

// =====================================================================
// Kernel 5: LSTM cell elementwise.
// =====================================================================
__global__ void lstm_elem_kernel(const float* __restrict__ gates,
                                 const float* __restrict__ b_hh,
                                 const float* __restrict__ query_g,
                                 float* __restrict__ c,
                                 float* __restrict__ h,
                                 int first) {
    int idx = blockIdx.x * blockDim.x + threadIdx.x;
    if (idx >= Bq * Hh) return;
    int n = idx / Hh, j = idx % Hh;
    const float* gb = gates + (size_t)n * NG;
    float gi = gb[j], gf = gb[Hh + j], gg = gb[2 * Hh + j], go = gb[3 * Hh + j];
    if (first) {
        gi += b_hh[j]; gf += b_hh[Hh + j]; gg += b_hh[2 * Hh + j]; go += b_hh[3 * Hh + j];
    }
    float cp = first ? 0.f : c[idx];
    float si = 1.f / (1.f + expf(-gi));
    float sf = 1.f / (1.f + expf(-gf));
    float sq = 1.f / (1.f + expf(-go));
    float cn = sf * cp + si * tanhf(gg);
    c[idx] = cn;
    float hr = sq * tanhf(cn);
    if (j < DM) h[(size_t)n * DM + j] = query_g[(size_t)n * DM + j] + hr;
}

// =====================================================================
// Kernel 6: out[n] = h[n] . support_g   (wave per row)
// =====================================================================
__global__ void final_dot_kernel(const float* __restrict__ h,
                                 const float* __restrict__ sg,
                                 float* __restrict__ out) {
    int row = blockIdx.x * blockDim.y + threadIdx.y;
    if (row >= Bq) return;
    int lane = threadIdx.x;
    const float* x = h + (size_t)row * DM;
    float s = 0.f;
    for (int j = lane; j < DM; j += 32) s += x[j] * sg[j];
    for (int off = 16; off; off >>= 1) s += __shfl_xor(s, off, 32);
    if (lane == 0) out[row] = s;
}

// =====================================================================
extern "C" void kernel_launch(void* const* d_in, const int* in_sizes, int n_in,
                              void* d_out, int out_size, void* d_ws, size_t ws_size,
                              hipStream_t stream) {
    const int*   qlc   = (const int*)d_in[2];
    const int*   qld   = (const int*)d_in[3];
    const int*   qrc   = (const int*)d_in[4];
    const int*   qrd   = (const int*)d_in[5];
    const int*   slc   = (const int*)d_in[6];
    const int*   sld   = (const int*)d_in[7];
    const int*   src_  = (const int*)d_in[8];
    const int*   srd   = (const int*)d_in[9];
    const float* emb   = (const float*)d_in[10];
    const float* gcn_w = (const float*)d_in[11];
    const float* gcn_b = (const float*)d_in[12];
    const float* p1w   = (const float*)d_in[13];
    const float* p1b   = (const float*)d_in[14];
    const float* p2w   = (const float*)d_in[15];
    const float* p2b   = (const float*)d_in[16];
    const float* lng   = (const float*)d_in[17];
    const float* lnb   = (const float*)d_in[18];
    const float* w_ih  = (const float*)d_in[19];
    const float* w_hh  = (const float*)d_in[20];
    const float* b_ih  = (const float*)d_in[21];
    const float* b_hh  = (const float*)d_in[22];
    float* out = (float*)d_out;

    float* ws = (float*)d_ws;
    size_t off = 0;
    auto alloc = [&](size_t n) { float* p = ws + off; off += n; return p; };
    float* Sq  = alloc((size_t)Bq * 2 * DM);
    float* Ss  = alloc((size_t)FEW * 2 * DM);
    float* Qn  = alloc((size_t)Bq * DM);
    float* Hb  = alloc((size_t)Bq * DI);
    float* Ob  = alloc((size_t)Bq * DM);
    float* Qg  = alloc((size_t)Bq * DM);
    float* Gih = alloc((size_t)Bq * NG);
    float* Gat = alloc((size_t)Bq * NG);
    float* Cst = alloc((size_t)Bq * Hh);
    float* Hst = alloc((size_t)Bq * DM);
    float* SG  = alloc(DM);
    float* AV  = alloc(NG);

    auto gemm = [&](const float* A, int lda, const float* W, int ldw,
                    const float* bias, float bscale, const float* res,
                    const int* deg, float* C, int ldc, int M, int N, int K, int act) {
        int blocks = (M / 64) * (N / 64);
        gemm_wmma_kernel<<<blocks, dim3(32, 8), 0, stream>>>(
            A, lda, W, ldw, bias, bscale, res, deg, C, ldc, M, N, K, act);
    };

    {
        int waves = (Bq + FEW) * 2;
        int grid  = (waves + 7) / 8;
        gather_sum_kernel<<<grid, 256, 0, stream>>>(qlc, qrc, slc, src_, emb, Sq, Ss);
    }
    support_path_kernel<<<1, 256, 0, stream>>>(Ss, sld, srd, gcn_w, gcn_b,
                                               p1w, p1b, p2w, p2b, lng, lnb,
                                               w_hh, b_hh, SG, AV);
    gemm(Sq,      2 * DM, gcn_w, DM, gcn_b, (float)NB, nullptr, qld, Qn,      DM, Bq, Dd, DM, 2);
    gemm(Sq + DM, 2 * DM, gcn_w, DM, gcn_b, (float)NB, nullptr, qrd, Qn + Dd, DM, Bq, Dd, DM, 2);
    gemm(Qn, DM, p1w, DM, p1b, 1.f, nullptr, nullptr, Hb, DI, Bq, DI, DM, 1); // relu
    gemm(Hb, DI, p2w, DI, p2b, 1.f, Qn,      nullptr, Ob, DM, Bq, DM, DI, 0); // +residual
    layernorm_kernel<<<Bq / 8, dim3(32, 8), 0, stream>>>(Ob, lng, lnb, Qg);
    gemm(Qg, DM, w_ih, DM, b_ih, 1.f, nullptr, nullptr, Gih, NG, Bq, NG, DM, 0);
    {
        int thr  = Bq * Hh;
        int grid = (thr + 255) / 256;
        lstm_elem_kernel<<<grid, 256, 0, stream>>>(Gih, b_hh, Qg, Cst, Hst, 1);
        for (int t = 1; t < NSTEPS; ++t) {
            gemm(Hst, DM, w_hh, Hh, AV, 1.f, Gih, nullptr, Gat, NG, Bq, NG, DM, 0);
            lstm_elem_kernel<<<grid, 256, 0, stream>>>(Gat, nullptr, Qg, Cst, Hst, 0);
        }
    }
    final_dot_kernel<<<Bq / 8, dim3(32, 8), 0, stream>>>(Hst, SG, out);
}